// ObjectCondensationLoss_15427522527888
// MI455X (gfx1250) — compile-verified
//
#include <hip/hip_runtime.h>
#include <hip/hip_bf16.h>

// Problem constants (fixed by the reference harness): B=16, N=65536, D=32, K=512.
#define OC_B 16
#define OC_K 512
#define OC_MARGIN 0.3f
#define HITS_PER_BLOCK 2048

typedef float v2f __attribute__((ext_vector_type(2)));
typedef float v8f __attribute__((ext_vector_type(8)));

// ---------------- init workspace ----------------
__global__ void oc_init(int* counts, int* cpcnt, int* ncpcnt, int* firstcp,
                        int* pos, float* segd2, float* segr, float* bce_sum,
                        float* repsum, int BK, int B, int N) {
    int i = blockIdx.x * blockDim.x + threadIdx.x;
    if (i < BK) {
        counts[i] = 0; cpcnt[i] = 0; ncpcnt[i] = 0; firstcp[i] = N;
        segd2[i] = 0.f; segr[i] = 0.f;
    }
    if (i < B) { pos[i] = 0; bce_sum[i] = 0.f; repsum[i] = 0.f; }
}

// ---------------- per-segment integer stats ----------------
__global__ void oc_stats(const int* __restrict__ sid, const int* __restrict__ iscp,
                         int* counts, int* cpcnt, int* ncpcnt, int* firstcp,
                         int* pos, int N, int K) {
    size_t gid = (size_t)blockIdx.x * blockDim.x + threadIdx.x;  // over B*N
    int b = (int)(gid / N);
    int n = (int)(gid % N);
    int s  = sid[gid];
    int cp = iscp[gid];
    if (cp) atomicAdd(&pos[b], 1);
    if (s >= 0) {
        int idx = b * K + s;
        atomicAdd(&counts[idx], 1);
        if (cp) { atomicAdd(&cpcnt[idx], 1); atomicMin(&firstcp[idx], n); }
        else    { atomicAdd(&ncpcnt[idx], 1); }
    }
}

// ---------------- gather cp embeddings / betas / norms ----------------
__global__ void oc_gather(const float* __restrict__ beta, const float* __restrict__ embed,
                          const int* __restrict__ firstcp, float* cpemb, float* cpbeta,
                          float* cpnorm, int N, int K, int D) {
    int i = blockIdx.x * blockDim.x + threadIdx.x;  // over B*K
    int b = i / K;
    int fc = firstcp[i];
    int safe = fc < N ? fc : (N - 1);   // reference's safe_cp = min(first_cp, N-1)
    const float4* src = (const float4*)(embed + ((size_t)b * N + safe) * D);
    float4* dst = (float4*)(cpemb + (size_t)i * D);
    float nrm = 0.f;
#pragma unroll
    for (int j = 0; j < 8; ++j) {  // D=32 -> 8 float4
        float4 v = src[j];
        dst[j] = v;
        nrm += v.x * v.x + v.y * v.y + v.z * v.z + v.w * v.w;
    }
    cpnorm[i] = nrm;
    cpbeta[i] = beta[(size_t)b * N + safe];
}

// ---------------- bandwidth-dominant streaming pass ----------------
// BCE + attraction d2 + ranking, LDS-staged segment accumulation.
__global__ void oc_main(const float* __restrict__ beta, const float* __restrict__ embed,
                        const int* __restrict__ sid, const int* __restrict__ iscp,
                        const float* __restrict__ cpemb, const float* __restrict__ cpbeta,
                        const int* __restrict__ pos, float* segd2, float* segr,
                        float* bce_sum, int N, int K, int D) {
    __shared__ float sd2[OC_K];
    __shared__ float srk[OC_K];
    __shared__ float red[256];

    int blocksPerBatch = N / HITS_PER_BLOCK;
    int b = blockIdx.x / blocksPerBatch;
    int base = (blockIdx.x % blocksPerBatch) * HITS_PER_BLOCK;

    for (int i = threadIdx.x; i < K; i += blockDim.x) { sd2[i] = 0.f; srk[i] = 0.f; }
    __syncthreads();

    float posf = (float)pos[b];
    float pw = ((float)N - posf) / (posf + 1e-6f);

    float bacc = 0.f;
    const size_t bb = (size_t)b * N;
    for (int it = 0; it < HITS_PER_BLOCK; it += blockDim.x) {
        int n = base + it + threadIdx.x;
        size_t g = bb + (size_t)n;
        int s   = sid[g];
        int cp  = iscp[g];
        float x = beta[g];
        bool valid = (s >= 0);
        float y = cp ? 1.f : 0.f;
        float w = cp ? pw : (valid ? 1.f : 2.f);
        float bce = fmaxf(x, 0.f) - x * y + log1pf(expf(-fabsf(x)));
        bacc += w * bce;
        if (valid) {
            const float4* e4 = (const float4*)(embed + g * (size_t)D);
            const float4* c4 = (const float4*)(cpemb + ((size_t)b * K + s) * (size_t)D);
            float d2 = 0.f;
#pragma unroll
            for (int j = 0; j < 8; ++j) {
                float4 e = e4[j];
                float4 c = c4[j];
                float dx = e.x - c.x, dy = e.y - c.y, dz = e.z - c.z, dw = e.w - c.w;
                d2 += dx * dx + dy * dy + dz * dz + dw * dw;
            }
            atomicAdd(&sd2[s], d2);               // ds_add_f32
            if (!cp) {
                float r = x + OC_MARGIN - cpbeta[b * K + s];
                if (r > 0.f) atomicAdd(&srk[s], r);
            }
        }
    }
    __syncthreads();
    for (int i = threadIdx.x; i < K; i += blockDim.x) {
        float v0 = sd2[i], v1 = srk[i];
        if (v0 != 0.f) atomicAdd(&segd2[b * K + i], v0);
        if (v1 != 0.f) atomicAdd(&segr[b * K + i], v1);
    }
    // reduce BCE within block
    red[threadIdx.x] = bacc;
    __syncthreads();
    for (int off = blockDim.x / 2; off > 0; off >>= 1) {
        if (threadIdx.x < off) red[threadIdx.x] += red[threadIdx.x + off];
        __syncthreads();
    }
    if (threadIdx.x == 0) atomicAdd(&bce_sum[b], red[0]);
}

// ---------------- repulsion: Gram matrix via V_WMMA_F32_16X16X4_F32 ----------------
// One wave per 16x16 tile of G = CP * CP^T, K-loop of 8 chained WMMAs (D=32).
// rep element = exp(-(|i|^2 + |j|^2 - 2*G_ij)).
__global__ void oc_rep(const float* __restrict__ cpemb, const float* __restrict__ cpnorm,
                       float* repsum, int K, int D) {
    int wid  = (int)((blockIdx.x * blockDim.x + threadIdx.x) >> 5);
    int lane = threadIdx.x & 31;
    int tpr = K / 16;                 // 32 tiles per row
    int tilesPerB = tpr * tpr;        // 1024
    int b  = wid / tilesPerB;
    int t  = wid % tilesPerB;
    int row0 = (t / tpr) * 16;
    int col0 = (t % tpr) * 16;
    int lm = lane & 15;
    int half = lane >> 4;

    const float* CP = cpemb + (size_t)b * K * D;
    v8f c = {0.f, 0.f, 0.f, 0.f, 0.f, 0.f, 0.f, 0.f};
    // A 16x4 f32 layout: lanes 0-15 hold (K+0,K+1), lanes 16-31 hold (K+2,K+3),
    // row M = lane%16. B = CP^T, so the B frag load is the same pattern at col0.
    for (int kk = 0; kk < D; kk += 4) {
        int ks = kk + 2 * half;
        v2f a, bb;
        a.x  = CP[(row0 + lm) * D + ks];
        a.y  = CP[(row0 + lm) * D + ks + 1];
        bb.x = CP[(col0 + lm) * D + ks];
        bb.y = CP[(col0 + lm) * D + ks + 1];
        c = __builtin_amdgcn_wmma_f32_16x16x4_f32(
                /*neg_a=*/false, a, /*neg_b=*/false, bb,
                /*c_mod=*/(short)0, c, /*reuse_a=*/false, /*reuse_b=*/false);
    }
    // C/D layout: lane L, vgpr r -> row = r + (L>=16)*8, col = L%16
    const float* NB = cpnorm + (size_t)b * K;
    float nc = NB[col0 + lm];
    float s = 0.f;
#pragma unroll
    for (int r = 0; r < 8; ++r) {
        float nr = NB[row0 + r + half * 8];
        s += __expf(2.f * c[r] - nr - nc);
    }
    for (int off = 16; off > 0; off >>= 1) s += __shfl_down(s, off, 32);
    if (lane == 0) atomicAdd(&repsum[b], s);
}

// ---------------- finalize: combine into 4 scalar outputs ----------------
__global__ void oc_final(const int* counts, const int* cpcnt, const int* ncpcnt,
                         const int* firstcp, const float* segd2, const float* segr,
                         const float* bce_sum, const float* repsum, float* out,
                         int B, int N, int K) {
    __shared__ float red[OC_K];
    int k = threadIdx.x;  // blockDim.x == K
    float sL = 0.f, sB = 0.f, sA = 0.f, sR = 0.f;
    for (int b = 0; b < B; ++b) {
        int idx = b * K + k;
        int cnt = counts[idx], cpc = cpcnt[idx], ncp = ncpcnt[idx], fc = firstcp[idx];
        bool exists = cnt > 0;
        bool hascp  = fc < N;
        float attr_k = (exists && hascp) ? segd2[idx] / (float)(cnt > 1 ? cnt : 1) : 0.f;
        float rank_k = (cpc == 1 && ncp > 0) ? segr[idx] / (float)(ncp > 1 ? ncp : 1) : 0.f;
        float uq = exists ? 1.f : 0.f;

        red[k] = attr_k; __syncthreads();
        for (int off = OC_K / 2; off > 0; off >>= 1) {
            if (k < off) red[k] += red[k + off];
            __syncthreads();
        }
        float attrSum = red[0]; __syncthreads();

        red[k] = rank_k; __syncthreads();
        for (int off = OC_K / 2; off > 0; off >>= 1) {
            if (k < off) red[k] += red[k + off];
            __syncthreads();
        }
        float rankSum = red[0]; __syncthreads();

        red[k] = uq; __syncthreads();
        for (int off = OC_K / 2; off > 0; off >>= 1) {
            if (k < off) red[k] += red[k + off];
            __syncthreads();
        }
        float uqSum = red[0]; __syncthreads();

        if (k == 0) {
            float ranking = rankSum / fmaxf(uqSum, 1.f);
            float bce = bce_sum[b] / (float)N;
            float bl = bce + 2.f * ranking;
            float rp = repsum[b] / ((float)K * (float)K);
            sL += bl + attrSum + rp;
            sB += bl;
            sA += attrSum;
            sR += rp;
        }
        __syncthreads();
    }
    if (k == 0) {
        float inv = 1.f / (float)B;
        out[0] = sL * inv;  // total loss mean
        out[1] = sB * inv;  // beta_loss mean
        out[2] = sA * inv;  // attraction mean
        out[3] = sR * inv;  // repulsion mean
    }
}

extern "C" void kernel_launch(void* const* d_in, const int* in_sizes, int n_in,
                              void* d_out, int out_size, void* d_ws, size_t ws_size,
                              hipStream_t stream) {
    const float* beta  = (const float*)d_in[0];  // [B,N,1]
    const float* embed = (const float*)d_in[1];  // [B,N,D]
    const int*   sid   = (const int*)d_in[2];    // [B,N]
    const int*   iscp  = (const int*)d_in[3];    // [B,N]
    float* out = (float*)d_out;

    const int B = OC_B;
    const int BN = in_sizes[2];
    const int N = BN / B;
    const int D = in_sizes[1] / in_sizes[0];
    const int K = OC_K;
    const int BK = B * K;

    // carve workspace (16-byte-friendly order)
    char* p = (char*)d_ws;
    float* cpemb  = (float*)p; p += (size_t)BK * D * sizeof(float);
    float* cpbeta = (float*)p; p += (size_t)BK * sizeof(float);
    float* cpnorm = (float*)p; p += (size_t)BK * sizeof(float);
    float* segd2  = (float*)p; p += (size_t)BK * sizeof(float);
    float* segr   = (float*)p; p += (size_t)BK * sizeof(float);
    int* counts   = (int*)p;   p += (size_t)BK * sizeof(int);
    int* cpcnt    = (int*)p;   p += (size_t)BK * sizeof(int);
    int* ncpcnt   = (int*)p;   p += (size_t)BK * sizeof(int);
    int* firstcp  = (int*)p;   p += (size_t)BK * sizeof(int);
    int* pos      = (int*)p;   p += (size_t)B * sizeof(int);
    float* bce_sum = (float*)p; p += (size_t)B * sizeof(float);
    float* repsum  = (float*)p; p += (size_t)B * sizeof(float);

    oc_init<<<(BK + 255) / 256, 256, 0, stream>>>(counts, cpcnt, ncpcnt, firstcp,
                                                  pos, segd2, segr, bce_sum, repsum,
                                                  BK, B, N);
    oc_stats<<<BN / 256, 256, 0, stream>>>(sid, iscp, counts, cpcnt, ncpcnt,
                                           firstcp, pos, N, K);
    oc_gather<<<BK / 256, 256, 0, stream>>>(beta, embed, firstcp, cpemb, cpbeta,
                                            cpnorm, N, K, D);
    oc_main<<<B * (N / HITS_PER_BLOCK), 256, 0, stream>>>(beta, embed, sid, iscp,
                                                          cpemb, cpbeta, pos,
                                                          segd2, segr, bce_sum,
                                                          N, K, D);
    // one wave per 16x16 tile: B * (K/16)^2 waves * 32 lanes
    int repThreads = B * (K / 16) * (K / 16) * 32;
    oc_rep<<<repThreads / 256, 256, 0, stream>>>(cpemb, cpnorm, repsum, K, D);
    oc_final<<<1, K, 0, stream>>>(counts, cpcnt, ncpcnt, firstcp, segd2, segr,
                                  bce_sum, repsum, out, B, N, K);
}